// SimpleGNN_65987877536207
// MI455X (gfx1250) — compile-verified
//
#include <hip/hip_runtime.h>
#include <hip/hip_bf16.h>
#include <math.h>

#define N_NODES 100000
#define N_EDGES 1600000
#define HDIM    128
#define NGRAPH  128
#define COUT    10
#define EPSV    1e-5f

typedef float v2f __attribute__((ext_vector_type(2)));
typedef float v8f __attribute__((ext_vector_type(8)));

__device__ __forceinline__ void atomAddF(float* p, float v) {
  unsafeAtomicAdd(p, v);   // native global_atomic_add_f32 on gfx1250
}

// ---------------- utility fills ----------------
__global__ void fill_val_k(float* __restrict__ p, float v, int n) {
  int t = blockIdx.x * blockDim.x + threadIdx.x;
  if (t < n) p[t] = v;
}

__global__ void fill_bias_rows_k(float* __restrict__ out, const float* __restrict__ bias, int nrows) {
  int t = blockIdx.x * blockDim.x + threadIdx.x;
  if (t < nrows * HDIM) out[t] = bias[t & (HDIM - 1)];
}

// ---------------- gcn norm: degree -> rsqrt ----------------
__global__ void deg_accum_k(const int* __restrict__ ei, float* __restrict__ deg) {
  int e = blockIdx.x * blockDim.x + threadIdx.x;
  if (e < N_EDGES) atomAddF(&deg[ei[N_EDGES + e]], 1.0f);  // dst row of edge_index
}

__global__ void deg_rsqrt_k(float* __restrict__ deg) {
  int i = blockIdx.x * blockDim.x + threadIdx.x;
  if (i < N_NODES) deg[i] = rsqrtf(deg[i]);  // deg >= 1 (self loop)
}

__global__ void cnt_accum_k(const int* __restrict__ batch, float* __restrict__ cnt) {
  int i = blockIdx.x * blockDim.x + threadIdx.x;
  if (i < N_NODES) atomAddF(&cnt[batch[i]], 1.0f);
}

__global__ void cnt_clamp_k(float* __restrict__ cnt) {
  int g = threadIdx.x;
  if (g < NGRAPH) cnt[g] = fmaxf(cnt[g], 1.0f);
}

// ---------------- fp32 WMMA GEMM: Out[nrows x 128] = X[nrows x 128] @ W[128 x 128] ----------------
// one wave computes a 16-row x 128-col strip via 8 accumulators of v_wmma_f32_16x16x4_f32
__global__ void gemm128_wmma_k(const float* __restrict__ X, const float* __restrict__ W,
                               float* __restrict__ Out, int nrows) {
  __shared__ float Wl[HDIM * HDIM];
  for (int i = threadIdx.x; i < HDIM * HDIM; i += blockDim.x) Wl[i] = W[i];
  __syncthreads();

  const int wave = threadIdx.x >> 5;
  const int lane = threadIdx.x & 31;
  const int half = lane >> 4;      // lanes 16..31 hold the second K pair / upper M rows
  const int l16  = lane & 15;
  const int strip = blockIdx.x * (blockDim.x >> 5) + wave;
  if (strip * 16 >= nrows) return;               // wave-uniform: EXEC all-ones inside WMMA

  const float* xrow = X + (size_t)(strip * 16 + l16) * HDIM;
  v8f acc[8] = {};

  for (int k0 = 0; k0 < HDIM; k0 += 4) {
    // A 16x4 frag: VGPR v holds K = v + 2*half  (rows M = l16)
    v2f a;
    a.x = xrow[k0 + 2 * half + 0];
    a.y = xrow[k0 + 2 * half + 1];
#pragma unroll
    for (int t = 0; t < 8; ++t) {
      // B 4x16 frag: VGPR v holds K = v + 2*half, N = t*16 + l16
      v2f b;
      b.x = Wl[(k0 + 2 * half + 0) * HDIM + t * 16 + l16];
      b.y = Wl[(k0 + 2 * half + 1) * HDIM + t * 16 + l16];
      acc[t] = __builtin_amdgcn_wmma_f32_16x16x4_f32(
          false, a, false, b, (short)0, acc[t], false, false);
    }
  }

  // C/D layout: VGPR v -> row M = v + 8*half, col N = t*16 + l16
  float* orow = Out + (size_t)strip * 16 * HDIM;
#pragma unroll
  for (int t = 0; t < 8; ++t)
#pragma unroll
    for (int v = 0; v < 8; ++v)
      orow[(size_t)(v + 8 * half) * HDIM + t * 16 + l16] = acc[t][v];
}

// ---------------- edge gather/scale/scatter (incl. implicit self loops) ----------------
__global__ void scatter_edges_k(const float* __restrict__ Hm, const int* __restrict__ ei,
                                const float* __restrict__ dinv, float* __restrict__ Out) {
  const int wid  = blockIdx.x * (blockDim.x >> 5) + (threadIdx.x >> 5);
  const int lane = threadIdx.x & 31;
  const int total = N_EDGES + N_NODES;
  if (wid >= total) return;
  int s, d;
  if (wid < N_EDGES) { s = ei[wid]; d = ei[N_EDGES + wid]; }
  else               { s = d = wid - N_EDGES; }           // self loop
  const float w = dinv[s] * dinv[d];
  const float4 v = ((const float4*)(Hm + (size_t)s * HDIM))[lane];
  float* op = Out + (size_t)d * HDIM + lane * 4;
  atomAddF(op + 0, v.x * w);
  atomAddF(op + 1, v.y * w);
  atomAddF(op + 2, v.z * w);
  atomAddF(op + 3, v.w * w);
}

// ---------------- graph-wise segment sums ----------------
__global__ void seg_accum_k(const float* __restrict__ Hm, const int* __restrict__ batch,
                            float* __restrict__ acc) {
  int t = blockIdx.x * blockDim.x + threadIdx.x;
  int node = t >> 5, q = t & 31;
  if (node >= N_NODES) return;
  int g = batch[node];
  float4 v = ((const float4*)(Hm + (size_t)node * HDIM))[q];
  float* a = acc + (size_t)g * HDIM + q * 4;
  atomAddF(a + 0, v.x); atomAddF(a + 1, v.y);
  atomAddF(a + 2, v.z); atomAddF(a + 3, v.w);
}

__global__ void mean_div_k(float* __restrict__ mean, const float* __restrict__ cnt) {
  int t = blockIdx.x * blockDim.x + threadIdx.x;
  if (t < NGRAPH * HDIM) mean[t] /= cnt[t >> 7];
}

__global__ void var_accum_k(const float* __restrict__ Hm, const int* __restrict__ batch,
                            const float* __restrict__ mean, const float* __restrict__ alpha,
                            float* __restrict__ var) {
  int t = blockIdx.x * blockDim.x + threadIdx.x;
  int node = t >> 5, q = t & 31;
  if (node >= N_NODES) return;
  int g = batch[node];
  int f = q * 4;
  float4 v = ((const float4*)(Hm + (size_t)node * HDIM))[q];
  const float* mrow = mean + (size_t)g * HDIM + f;
  float s0 = v.x - alpha[f + 0] * mrow[0];
  float s1 = v.y - alpha[f + 1] * mrow[1];
  float s2 = v.z - alpha[f + 2] * mrow[2];
  float s3 = v.w - alpha[f + 3] * mrow[3];
  float* a = var + (size_t)g * HDIM + f;
  atomAddF(a + 0, s0 * s0); atomAddF(a + 1, s1 * s1);
  atomAddF(a + 2, s2 * s2); atomAddF(a + 3, s3 * s3);
}

__global__ void gn_apply_k(const float* __restrict__ Hm, const int* __restrict__ batch,
                           const float* __restrict__ mean, const float* __restrict__ var,
                           const float* __restrict__ cnt,
                           const float* __restrict__ gw, const float* __restrict__ gb,
                           const float* __restrict__ alpha, float* __restrict__ Out) {
  int t = blockIdx.x * blockDim.x + threadIdx.x;
  int node = t >> 5, q = t & 31;
  if (node >= N_NODES) return;
  int g = batch[node];
  int f = q * 4;
  float inv = 1.0f / cnt[g];
  float4 v = ((const float4*)(Hm + (size_t)node * HDIM))[q];
  const float* mrow = mean + (size_t)g * HDIM + f;
  const float* vrow = var + (size_t)g * HDIM + f;
  float* po = Out + (size_t)node * HDIM + f;
  float vals[4] = {v.x, v.y, v.z, v.w};
#pragma unroll
  for (int i = 0; i < 4; ++i) {
    float sub = vals[i] - alpha[f + i] * mrow[i];
    float r   = rsqrtf(vrow[i] * inv + EPSV);
    float y   = gw[f + i] * sub * r + gb[f + i];
    po[i] = fmaxf(y, 0.0f);
  }
}

// ---------------- head: mean pool -> linear -> log_softmax ----------------
__global__ void head_k(const float* __restrict__ pooled, const float* __restrict__ cnt,
                       const float* __restrict__ lw, const float* __restrict__ lb,
                       float* __restrict__ out) {
  int g = threadIdx.x;
  if (g >= NGRAPH) return;
  float inv = 1.0f / cnt[g];
  float logits[COUT];
#pragma unroll
  for (int c = 0; c < COUT; ++c) logits[c] = lb[c];
  for (int k = 0; k < HDIM; ++k) {
    float p = pooled[(size_t)g * HDIM + k] * inv;
#pragma unroll
    for (int c = 0; c < COUT; ++c) logits[c] += p * lw[k * COUT + c];
  }
  float m = logits[0];
#pragma unroll
  for (int c = 1; c < COUT; ++c) m = fmaxf(m, logits[c]);
  float s = 0.0f;
#pragma unroll
  for (int c = 0; c < COUT; ++c) s += expf(logits[c] - m);
  float lse = logf(s);
#pragma unroll
  for (int c = 0; c < COUT; ++c) out[g * COUT + c] = logits[c] - m - lse;
}

// ---------------- driver ----------------
extern "C" void kernel_launch(void* const* d_in, const int* in_sizes, int n_in,
                              void* d_out, int out_size, void* d_ws, size_t ws_size,
                              hipStream_t stream) {
  const float* x   = (const float*)d_in[0];
  const int*   ei  = (const int*)d_in[1];     // [2, E]: row0 src, row1 dst
  const int*   bat = (const int*)d_in[2];
  const float* w1  = (const float*)d_in[3];  const float* b1 = (const float*)d_in[4];
  const float* w2  = (const float*)d_in[5];  const float* b2 = (const float*)d_in[6];
  const float* w3  = (const float*)d_in[7];  const float* b3 = (const float*)d_in[8];
  const float* g1w = (const float*)d_in[9];  const float* g1b = (const float*)d_in[10]; const float* g1a = (const float*)d_in[11];
  const float* g2w = (const float*)d_in[12]; const float* g2b = (const float*)d_in[13]; const float* g2a = (const float*)d_in[14];
  const float* g3w = (const float*)d_in[15]; const float* g3b = (const float*)d_in[16]; const float* g3a = (const float*)d_in[17];
  const float* lw  = (const float*)d_in[18]; const float* lb = (const float*)d_in[19];
  float* out = (float*)d_out;

  float* ws     = (float*)d_ws;
  float* dinv   = ws;                              // N
  float* cnt    = dinv + N_NODES;                  // G
  float* hA     = cnt + NGRAPH;                    // N*128
  float* hB     = hA + (size_t)N_NODES * HDIM;     // N*128
  float* mean   = hB + (size_t)N_NODES * HDIM;     // G*128
  float* var    = mean + NGRAPH * HDIM;            // G*128
  float* pooled = var + NGRAPH * HDIM;             // G*128

  const int TB = 256;
  const int gN    = (N_NODES + TB - 1) / TB;
  const int gE    = (N_EDGES + TB - 1) / TB;
  const int gNH   = ((size_t)N_NODES * HDIM + TB - 1) / TB;     // 50000
  const int gGH   = (NGRAPH * HDIM + TB - 1) / TB;              // 64
  const int gNW   = ((size_t)N_NODES * 32 + TB - 1) / TB;       // node-per-wave kernels
  const int gScat = ((size_t)(N_EDGES + N_NODES) * 32 + TB - 1) / TB;
  const int gGemm = ((N_NODES / 16) + 7) / 8;                   // 8 strips/block

  // normalization terms
  fill_val_k<<<gN, TB, 0, stream>>>(dinv, 1.0f, N_NODES);       // self-loop degree
  deg_accum_k<<<gE, TB, 0, stream>>>(ei, dinv);
  deg_rsqrt_k<<<gN, TB, 0, stream>>>(dinv);
  fill_val_k<<<1, TB, 0, stream>>>(cnt, 0.0f, NGRAPH);
  cnt_accum_k<<<gN, TB, 0, stream>>>(bat, cnt);
  cnt_clamp_k<<<1, NGRAPH, 0, stream>>>(cnt);

  // ---- layer 1: x -> hA(gemm) ; hB = b1 + scatter(hA) ; gn -> hA
  gemm128_wmma_k<<<gGemm, TB, 0, stream>>>(x, w1, hA, N_NODES);
  fill_bias_rows_k<<<gNH, TB, 0, stream>>>(hB, b1, N_NODES);
  scatter_edges_k<<<gScat, TB, 0, stream>>>(hA, ei, dinv, hB);
  fill_val_k<<<gGH, TB, 0, stream>>>(mean, 0.0f, NGRAPH * HDIM);
  seg_accum_k<<<gNW, TB, 0, stream>>>(hB, bat, mean);
  mean_div_k<<<gGH, TB, 0, stream>>>(mean, cnt);
  fill_val_k<<<gGH, TB, 0, stream>>>(var, 0.0f, NGRAPH * HDIM);
  var_accum_k<<<gNW, TB, 0, stream>>>(hB, bat, mean, g1a, var);
  gn_apply_k<<<gNW, TB, 0, stream>>>(hB, bat, mean, var, cnt, g1w, g1b, g1a, hA);

  // ---- layer 2: hA -> hB(gemm) ; hA = b2 + scatter(hB) ; gn -> hB
  gemm128_wmma_k<<<gGemm, TB, 0, stream>>>(hA, w2, hB, N_NODES);
  fill_bias_rows_k<<<gNH, TB, 0, stream>>>(hA, b2, N_NODES);
  scatter_edges_k<<<gScat, TB, 0, stream>>>(hB, ei, dinv, hA);
  fill_val_k<<<gGH, TB, 0, stream>>>(mean, 0.0f, NGRAPH * HDIM);
  seg_accum_k<<<gNW, TB, 0, stream>>>(hA, bat, mean);
  mean_div_k<<<gGH, TB, 0, stream>>>(mean, cnt);
  fill_val_k<<<gGH, TB, 0, stream>>>(var, 0.0f, NGRAPH * HDIM);
  var_accum_k<<<gNW, TB, 0, stream>>>(hA, bat, mean, g2a, var);
  gn_apply_k<<<gNW, TB, 0, stream>>>(hA, bat, mean, var, cnt, g2w, g2b, g2a, hB);

  // ---- layer 3: hB -> hA(gemm) ; hB = b3 + scatter(hA) ; gn -> hA
  gemm128_wmma_k<<<gGemm, TB, 0, stream>>>(hB, w3, hA, N_NODES);
  fill_bias_rows_k<<<gNH, TB, 0, stream>>>(hB, b3, N_NODES);
  scatter_edges_k<<<gScat, TB, 0, stream>>>(hA, ei, dinv, hB);
  fill_val_k<<<gGH, TB, 0, stream>>>(mean, 0.0f, NGRAPH * HDIM);
  seg_accum_k<<<gNW, TB, 0, stream>>>(hB, bat, mean);
  mean_div_k<<<gGH, TB, 0, stream>>>(mean, cnt);
  fill_val_k<<<gGH, TB, 0, stream>>>(var, 0.0f, NGRAPH * HDIM);
  var_accum_k<<<gNW, TB, 0, stream>>>(hB, bat, mean, g3a, var);
  gn_apply_k<<<gNW, TB, 0, stream>>>(hB, bat, mean, var, cnt, g3w, g3b, g3a, hA);

  // ---- pool + head
  fill_val_k<<<gGH, TB, 0, stream>>>(pooled, 0.0f, NGRAPH * HDIM);
  seg_accum_k<<<gNW, TB, 0, stream>>>(hA, bat, pooled);
  head_k<<<1, NGRAPH, 0, stream>>>(pooled, cnt, lw, lb, out);
}